// RGAT_80410377716214
// MI455X (gfx1250) — compile-verified
//
#include <hip/hip_runtime.h>
#include <hip/hip_bf16.h>
#include <math.h>

#define D_DIM 200
#define N_NODES 20000
#define E_EDGES 320000
#define R_REL 512
#define DQ (D_DIM / 4)   // 50 float4 per row

typedef float v2f __attribute__((ext_vector_type(2)));
typedef float v8f __attribute__((ext_vector_type(8)));

// ---------------------------------------------------------------------------
// WMMA fp32 GEMM:  C[M x Dout] = A[M x K] @ B[K x Dout]
// One wave32 computes one 16x16 tile of C via V_WMMA_F32_16X16X4_F32.
// M % 16 == 0, K % 4 == 0; Dout guarded via clamped loads + cndmask (no EXEC
// manipulation inside the k-loop -> WMMA sees EXEC == all-1s, clean codegen).
// ---------------------------------------------------------------------------
__global__ void wmma_gemm_f32(const float* __restrict__ A,
                              const float* __restrict__ B,
                              float* __restrict__ C,
                              int M, int K, int Dout) {
    const int lane   = threadIdx.x & 31;
    const int waveIb = threadIdx.x >> 5;
    const int wavesB = blockDim.x >> 5;
    const int tile   = blockIdx.x * wavesB + waveIb;
    const int ntiles = (Dout + 15) >> 4;
    const int mtiles = M >> 4;
    if (tile >= mtiles * ntiles) return;          // wave-uniform exit

    const int mt = tile / ntiles;
    const int nt = tile - mt * ntiles;
    const int m0 = mt << 4;
    const int n0 = nt << 4;

    const int half = lane >> 4;                   // 0 or 1
    const int l15  = lane & 15;
    const int arow = m0 + l15;
    const int bcol = n0 + l15;
    const bool bok = (bcol < Dout);
    const int  bc  = bok ? bcol : 0;              // clamped, always-valid addr
    const float bmask = bok ? 1.0f : 0.0f;

    const float* Arow = A + (size_t)arow * K + half * 2;  // covers kk, kk+1
    const float* Bcol = B + bc;

    v8f acc = {0.f, 0.f, 0.f, 0.f, 0.f, 0.f, 0.f, 0.f};

    for (int k = 0; k < K; k += 4) {
        const int kk = k + half * 2;
        v2f a, b;
        a.x = Arow[k];                             // contiguous pair -> b64
        a.y = Arow[k + 1];
        b.x = Bcol[(size_t)kk * Dout] * bmask;     // unconditional loads
        b.y = Bcol[(size_t)(kk + 1) * Dout] * bmask;
        acc = __builtin_amdgcn_wmma_f32_16x16x4_f32(
            /*neg_a=*/false, a, /*neg_b=*/false, b,
            /*c_mod=*/(short)0, acc, /*reuse_a=*/false, /*reuse_b=*/false);
    }

    if (bok) {
#pragma unroll
        for (int v = 0; v < 8; ++v) {
            const int mrow = m0 + v + half * 8;
            C[(size_t)mrow * Dout + bcol] = acc[v];
        }
    }
}

// colsumQ[d] = sum_k w_quad[k][d]   (rank-1 term for "+ fre[:,None]")
__global__ void colsum_kernel(const float* __restrict__ wq,
                              float* __restrict__ colsum, int K, int D) {
    int d = blockIdx.x * blockDim.x + threadIdx.x;
    if (d >= D) return;
    float s = 0.f;
    for (int k = 0; k < K; ++k) s += wq[(size_t)k * D + d];
    colsum[d] = s;
}

__device__ __forceinline__ unsigned enc_f32(float f) {
    unsigned u = __float_as_uint(f);
    return (u & 0x80000000u) ? ~u : (u | 0x80000000u);
}
__device__ __forceinline__ float dec_f32(unsigned u) {
    return (u & 0x80000000u) ? __uint_as_float(u ^ 0x80000000u)
                             : __uint_as_float(~u);
}
__device__ __forceinline__ float lrelu(float a) {
    return (a > 0.f) ? a : 0.01f * a;
}

// Pass 1: per-edge logit -> atomic segment max (ordered-uint), + degree.
// One wave32 per edge; lanes stride over 50 float4 slots (b128 loads).
__global__ void edge_pass1(const int* __restrict__ esrc, const int* __restrict__ edst,
                           const int* __restrict__ etyp, const float* __restrict__ fre,
                           const float4* __restrict__ Qs, const float4* __restrict__ Qd,
                           const float4* __restrict__ Qr, const float4* __restrict__ colsumQ,
                           unsigned* __restrict__ segmax_u, float* __restrict__ deg,
                           int E) {
    const int wave = (int)((blockIdx.x * blockDim.x + threadIdx.x) >> 5);
    const int lane = threadIdx.x & 31;
    if (wave >= E) return;
    const int s = esrc[wave], t = edst[wave], r = etyp[wave];
    const float f = fre[wave];
    if (lane == 0) atomicAdd(&deg[t], 1.0f);
    const float4* qs = Qs + (size_t)s * DQ;
    const float4* qd = Qd + (size_t)t * DQ;
    const float4* qr = Qr + (size_t)r * DQ;
    unsigned* sm = segmax_u + (size_t)t * D_DIM;
    for (int q = lane; q < DQ; q += 32) {
        float4 a = qs[q], b = qr[q], c = qd[q], w = colsumQ[q];
        float a0 = lrelu(a.x + b.x + c.x + f * w.x);
        float a1 = lrelu(a.y + b.y + c.y + f * w.y);
        float a2 = lrelu(a.z + b.z + c.z + f * w.z);
        float a3 = lrelu(a.w + b.w + c.w + f * w.w);
        atomicMax(&sm[4 * q + 0], enc_f32(a0));
        atomicMax(&sm[4 * q + 1], enc_f32(a1));
        atomicMax(&sm[4 * q + 2], enc_f32(a2));
        atomicMax(&sm[4 * q + 3], enc_f32(a3));
    }
}

// Pass 2: recompute logit + triplet, accumulate exp-sum and exp-weighted sum.
__global__ void edge_pass2(const int* __restrict__ esrc, const int* __restrict__ edst,
                           const int* __restrict__ etyp, const float* __restrict__ fre,
                           const float4* __restrict__ Ps, const float4* __restrict__ Pd,
                           const float4* __restrict__ Pr,
                           const float4* __restrict__ Qs, const float4* __restrict__ Qd,
                           const float4* __restrict__ Qr, const float4* __restrict__ colsumQ,
                           const unsigned* __restrict__ segmax_u,
                           float* __restrict__ segsum, float* __restrict__ hnum,
                           int E) {
    const int wave = (int)((blockIdx.x * blockDim.x + threadIdx.x) >> 5);
    const int lane = threadIdx.x & 31;
    if (wave >= E) return;
    const int s = esrc[wave], t = edst[wave], r = etyp[wave];
    const float f = fre[wave];
    const float4* ps = Ps + (size_t)s * DQ;
    const float4* pd = Pd + (size_t)t * DQ;
    const float4* pr = Pr + (size_t)r * DQ;
    const float4* qs = Qs + (size_t)s * DQ;
    const float4* qd = Qd + (size_t)t * DQ;
    const float4* qr = Qr + (size_t)r * DQ;
    const unsigned* sm = segmax_u + (size_t)t * D_DIM;
    float* ss = segsum + (size_t)t * D_DIM;
    float* hn = hnum + (size_t)t * D_DIM;
    for (int q = lane; q < DQ; q += 32) {
        float4 a = qs[q], b = qr[q], c = qd[q], w = colsumQ[q];
        float4 x = ps[q], y = pr[q], z = pd[q];
        float t0 = x.x + y.x + z.x;
        float t1 = x.y + y.y + z.y;
        float t2 = x.z + y.z + z.z;
        float t3 = x.w + y.w + z.w;
        float e0 = __expf(lrelu(a.x + b.x + c.x + f * w.x) - dec_f32(sm[4 * q + 0]));
        float e1 = __expf(lrelu(a.y + b.y + c.y + f * w.y) - dec_f32(sm[4 * q + 1]));
        float e2 = __expf(lrelu(a.z + b.z + c.z + f * w.z) - dec_f32(sm[4 * q + 2]));
        float e3 = __expf(lrelu(a.w + b.w + c.w + f * w.w) - dec_f32(sm[4 * q + 3]));
        atomicAdd(&ss[4 * q + 0], e0);
        atomicAdd(&ss[4 * q + 1], e1);
        atomicAdd(&ss[4 * q + 2], e2);
        atomicAdd(&ss[4 * q + 3], e3);
        atomicAdd(&hn[4 * q + 0], e0 * t0);
        atomicAdd(&hn[4 * q + 1], e1 * t1);
        atomicAdd(&hn[4 * q + 2], e2 * t2);
        atomicAdd(&hn[4 * q + 3], e3 * t3);
    }
}

// Final: out = relu( (deg>0 ? hnum/segsum*norm : 0) + (deg>0 ? L : L2) )
// float4-vectorized: 50 float4s per node row, so a float4 never crosses rows.
__global__ void finalize_kernel(const float4* __restrict__ hnum,
                                const float4* __restrict__ segsum,
                                const float* __restrict__ deg,
                                const float* __restrict__ norm,
                                const float4* __restrict__ L,
                                const float4* __restrict__ L2,
                                float4* __restrict__ out, int N) {
    size_t q = (size_t)blockIdx.x * blockDim.x + threadIdx.x;
    if (q >= (size_t)N * DQ) return;
    int node = (int)(q / DQ);
    bool has = deg[node] > 0.f;
    float nrm = has ? norm[node] : 0.f;
    float4 hv = hnum[q], sv = segsum[q];
    float4 lv = has ? L[q] : L2[q];
    float4 o;
    o.x = fmaxf((has ? (hv.x / sv.x) * nrm : 0.f) + lv.x, 0.f);
    o.y = fmaxf((has ? (hv.y / sv.y) * nrm : 0.f) + lv.y, 0.f);
    o.z = fmaxf((has ? (hv.z / sv.z) * nrm : 0.f) + lv.z, 0.f);
    o.w = fmaxf((has ? (hv.w / sv.w) * nrm : 0.f) + lv.w, 0.f);
    out[q] = o;
}

extern "C" void kernel_launch(void* const* d_in, const int* in_sizes, int n_in,
                              void* d_out, int out_size, void* d_ws, size_t ws_size,
                              hipStream_t stream) {
    (void)in_sizes; (void)n_in; (void)out_size; (void)ws_size;
    const int N = N_NODES, D = D_DIM, E = E_EDGES, R = R_REL;

    const float* node   = (const float*)d_in[0];
    const float* rel    = (const float*)d_in[1];
    const float* norm   = (const float*)d_in[2];
    const float* fre    = (const float*)d_in[3];
    const float* w_trip = (const float*)d_in[4];   // (3D, D)
    const float* w_quad = (const float*)d_in[5];   // (D, D)
    const float* w_loop = (const float*)d_in[6];
    const float* w_evlp = (const float*)d_in[7];
    const int*   esrc   = (const int*)d_in[8];
    const int*   edst   = (const int*)d_in[9];
    const int*   etyp   = (const int*)d_in[10];

    const float* W_s = w_trip;                    // rows [0,   D)
    const float* W_r = w_trip + (size_t)D * D;    // rows [D,  2D)
    const float* W_d = w_trip + (size_t)2 * D * D;// rows [2D, 3D)

    // ---- workspace carve-up (floats); peak ~113 MB ----
    const size_t ND = (size_t)N * D;              // 4,000,000
    const size_t RD = (size_t)R * D;              // 102,400
    float*    Ps      = (float*)d_ws;             // [ND]
    float*    Pd      = Ps + ND;                  // [ND]
    float*    Qs      = Pd + ND;                  // [ND]  (reused as L)
    float*    Qd      = Qs + ND;                  // [ND]  (reused as L2)
    unsigned* segmaxU = (unsigned*)(Qd + ND);     // [ND]
    float*    segsum  = (float*)(segmaxU + ND);   // [ND]
    float*    hnum    = segsum + ND;              // [ND]
    float*    Pr      = hnum + ND;                // [RD]
    float*    Qr      = Pr + RD;                  // [RD]
    float*    colsumQ = Qr + RD;                  // [D]
    float*    deg     = colsumQ + D;              // [N]

    const int THREADS = 256;                      // 8 wave32s
    const int wavesPerBlock = THREADS / 32;
    const int ntiles = (D + 15) / 16;             // 13
    auto gemmBlocks = [&](int M) {
        int tiles = (M / 16) * ntiles;
        return (tiles + wavesPerBlock - 1) / wavesPerBlock;
    };

    // ---- dense precompute (fp32 WMMA) ----
    wmma_gemm_f32<<<gemmBlocks(N), THREADS, 0, stream>>>(node, W_s, Ps, N, D, D);
    wmma_gemm_f32<<<gemmBlocks(N), THREADS, 0, stream>>>(node, W_d, Pd, N, D, D);
    wmma_gemm_f32<<<gemmBlocks(N), THREADS, 0, stream>>>(Ps, w_quad, Qs, N, D, D);
    wmma_gemm_f32<<<gemmBlocks(N), THREADS, 0, stream>>>(Pd, w_quad, Qd, N, D, D);
    wmma_gemm_f32<<<gemmBlocks(R), THREADS, 0, stream>>>(rel, W_r, Pr, R, D, D);
    wmma_gemm_f32<<<gemmBlocks(R), THREADS, 0, stream>>>(Pr, w_quad, Qr, R, D, D);
    colsum_kernel<<<1, 256, 0, stream>>>(w_quad, colsumQ, D, D);

    // ---- zero accumulators (graph-capture-safe async memsets) ----
    hipMemsetAsync(segmaxU, 0, ND * sizeof(unsigned), stream); // enc-domain -inf
    hipMemsetAsync(segsum,  0, ND * sizeof(float),    stream);
    hipMemsetAsync(hnum,    0, ND * sizeof(float),    stream);
    hipMemsetAsync(deg,     0, (size_t)N * sizeof(float), stream);

    // ---- edge passes: one wave per edge, float4 row traffic ----
    const int edgeBlocks = (E + wavesPerBlock - 1) / wavesPerBlock;
    edge_pass1<<<edgeBlocks, THREADS, 0, stream>>>(
        esrc, edst, etyp, fre,
        (const float4*)Qs, (const float4*)Qd, (const float4*)Qr,
        (const float4*)colsumQ, segmaxU, deg, E);
    edge_pass2<<<edgeBlocks, THREADS, 0, stream>>>(
        esrc, edst, etyp, fre,
        (const float4*)Ps, (const float4*)Pd, (const float4*)Pr,
        (const float4*)Qs, (const float4*)Qd, (const float4*)Qr,
        (const float4*)colsumQ, segmaxU, segsum, hnum, E);

    // ---- self-loop GEMMs (reuse Qs/Qd buffers; stream-ordered after pass2) ----
    float* L  = Qs;
    float* L2 = Qd;
    wmma_gemm_f32<<<gemmBlocks(N), THREADS, 0, stream>>>(node, w_loop, L,  N, D, D);
    wmma_gemm_f32<<<gemmBlocks(N), THREADS, 0, stream>>>(node, w_evlp, L2, N, D, D);

    // ---- finalize (float4) ----
    const size_t totalQ = (size_t)N * DQ;
    const int finBlocks = (int)((totalQ + THREADS - 1) / THREADS);
    finalize_kernel<<<finBlocks, THREADS, 0, stream>>>(
        (const float4*)hnum, (const float4*)segsum, deg, norm,
        (const float4*)L, (const float4*)L2, (float4*)d_out, N);
}